// SSIM_25692494364890
// MI455X (gfx1250) — compile-verified
//
#include <hip/hip_runtime.h>
#include <hip/hip_bf16.h>

typedef float v2f __attribute__((ext_vector_type(2)));
typedef float v8f __attribute__((ext_vector_type(8)));

#define WS      11
#define PAD     5
#define TILE    16
#define PROWS   26      // TILE + WS - 1
#define PCOLS   26
#define PSTRIDE 28      // patch row stride (floats): conflict-free strided A reads
#define KPAD    28      // K padded to multiple of 4 for 16x16x4 WMMA
#define NKC     7       // KPAD / 4
#define HROWS   28
#define HSTR    18      // H row stride: keeps store/load halves in disjoint banks
#define IMGDIM  512
#define NTX     (IMGDIM / TILE)   // 32 tiles per row
#define WPB     4       // waves per block

#define SSIM_C1 1.0e-4f
#define SSIM_C2 9.0e-4f

__global__ void ssim_zero_out(float* out) {
    if (threadIdx.x == 0 && blockIdx.x == 0) out[0] = 0.0f;
}

#define WMMA_F32(A, B, C) \
    __builtin_amdgcn_wmma_f32_16x16x4_f32(false, (A), false, (B), (short)0, (C), false, false)

__global__ __launch_bounds__(WPB * 32)
void ssim_wmma_kernel(const float* __restrict__ img1,
                      const float* __restrict__ img2,
                      float* __restrict__ out,
                      int n_tiles, float inv_total) {
    // Zero-padded Gaussian band table: s_band[d+16] = g[d] for d in [0,11), else 0.
    __shared__ float s_band[48];
    __shared__ float s_patch[WPB][2][PROWS * PSTRIDE];
    __shared__ float s_H[WPB][5][HROWS * HSTR];

    const int lane = threadIdx.x & 31;
    const int wave = threadIdx.x >> 5;

    if (threadIdx.x < 48) {
        int d = (int)threadIdx.x - 16;
        float v = 0.0f;
        if (d >= 0 && d < WS) {
            float x = (float)(d - PAD);
            float sum = 1.0f + 2.0f * (expf(-1.f) + expf(-4.f) + expf(-9.f) +
                                       expf(-16.f) + expf(-25.f));
            v = expf(-x * x) / sum;
        }
        s_band[threadIdx.x] = v;
    }
    __syncthreads();

    const int tile = blockIdx.x * WPB + wave;
    const bool active = (tile < n_tiles);
    const int t = active ? tile : 0;
    const int plane = t >> 10;
    const int ty    = (t & 1023) >> 5;
    const int tx    = t & 31;
    const int y0 = ty * TILE;
    const int x0 = tx * TILE;
    const size_t base = (size_t)plane * IMGDIM * IMGDIM;

    float* P1 = &s_patch[wave][0][0];
    float* P2 = &s_patch[wave][1][0];
    float* Hb = &s_H[wave][0][0];

    // ---- Load 26x26 halo patch (zero padded; cols 26..27 zeroed) ----
    const bool interior = active && (ty > 0) && (ty < NTX - 1) && (tx > 0) && (tx < NTX - 1);
    if (interior) {
        // Wave-uniform fast path: all addresses in-bounds, unconditional loads.
        const float* g1 = img1 + base + (size_t)(y0 - PAD) * IMGDIM + (x0 - PAD);
        const float* g2 = img2 + base + (size_t)(y0 - PAD) * IMGDIM + (x0 - PAD);
        if (lane < PSTRIDE) {
            const bool incol = (lane < PCOLS);
            for (int r = 0; r < PROWS; ++r) {
                float a = g1[r * IMGDIM + lane];
                float b = g2[r * IMGDIM + lane];
                P1[r * PSTRIDE + lane] = incol ? a : 0.0f;
                P2[r * PSTRIDE + lane] = incol ? b : 0.0f;
            }
        }
    } else {
        for (int r = 0; r < PROWS; ++r) {
            if (lane < PSTRIDE) {
                int gy = y0 + r - PAD;
                int gx = x0 + lane - PAD;
                bool ok = active && (lane < PCOLS) &&
                          (gy >= 0) && (gy < IMGDIM) && (gx >= 0) && (gx < IMGDIM);
                size_t idx = base + (size_t)gy * IMGDIM + gx;
                P1[r * PSTRIDE + lane] = ok ? img1[idx] : 0.0f;
                P2[r * PSTRIDE + lane] = ok ? img2[idx] : 0.0f;
            }
        }
    }

    // ---- Per-lane constant Gaussian band fragments (serve BOTH passes):
    // horizontal B-frag Gh[p][n] = g[p-n]  and  vertical A-frag Gv[m][p] = g[p-m]
    const int m    = lane & 15;         // A row (vert) / B col (horiz) / tile col
    const int koff = (lane >> 4) * 2;   // upper half-wave holds K+2,K+3 slices
    v2f gband[NKC];
#pragma unroll
    for (int kc = 0; kc < NKC; ++kc) {
        int p0 = kc * 4 + koff;
        gband[kc].x = s_band[p0 - m + 16];       // index in [1,43] -- always safe
        gband[kc].y = s_band[p0 + 1 - m + 16];
    }

    const v8f vzero = {0.f, 0.f, 0.f, 0.f, 0.f, 0.f, 0.f, 0.f};

    // ---- Horizontal pass via WMMA: H(32x16) = P(32x28) x Gh(28x16), 2 row chunks ----
#pragma unroll
    for (int rc = 0; rc < 2; ++rc) {
        int prow  = rc * 16 + m;                 // patch row this lane supplies to A
        int prowc = (prow > PROWS - 1) ? (PROWS - 1) : prow;  // clamped (safe addr)
        bool rok  = (prow < PROWS);
        const float* p1r = &P1[prowc * PSTRIDE];
        const float* p2r = &P2[prowc * PSTRIDE];

        v8f h1 = vzero, h2 = vzero, h11 = vzero, h22 = vzero, h12 = vzero;
#pragma unroll
        for (int kc = 0; kc < NKC; ++kc) {
            int kk = kc * 4 + koff;
            float v0 = p1r[kk];                  // unconditional loads, then cndmask
            float v1 = p1r[kk + 1];
            float w0 = p2r[kk];
            float w1 = p2r[kk + 1];
            v2f a1, a2, a11, a22, a12;
            a1.x = rok ? v0 : 0.0f;  a1.y = rok ? v1 : 0.0f;
            a2.x = rok ? w0 : 0.0f;  a2.y = rok ? w1 : 0.0f;
            a11.x = a1.x * a1.x;     a11.y = a1.y * a1.y;
            a22.x = a2.x * a2.x;     a22.y = a2.y * a2.y;
            a12.x = a1.x * a2.x;     a12.y = a1.y * a2.y;
            h1  = WMMA_F32(a1,  gband[kc], h1);
            h2  = WMMA_F32(a2,  gband[kc], h2);
            h11 = WMMA_F32(a11, gband[kc], h11);
            h22 = WMMA_F32(a22, gband[kc], h22);
            h12 = WMMA_F32(a12, gband[kc], h12);
        }
        // D layout: VGPR i -> row rc*16 + i (+8 for upper half-wave), col m.
        // Rows 26..31 are exact zeros (A rows were zeroed), so storing rows < 28
        // also initializes the vertical K-pad rows 26..27.
        int hbase = rc * 16 + ((lane >> 4) << 3);
#pragma unroll
        for (int i = 0; i < 8; ++i) {
            int hrow = hbase + i;
            if (hrow < KPAD) {
                int o = hrow * HSTR + m;
                Hb[0 * HROWS * HSTR + o] = h1[i];
                Hb[1 * HROWS * HSTR + o] = h2[i];
                Hb[2 * HROWS * HSTR + o] = h11[i];
                Hb[3 * HROWS * HSTR + o] = h22[i];
                Hb[4 * HROWS * HSTR + o] = h12[i];
            }
        }
    }

    // ---- Vertical pass via WMMA: Out(16x16) = Gv(16x28) x H(28x16) ----
    v8f c1 = vzero, c2 = vzero, c11 = vzero, c22 = vzero, c12 = vzero;
#pragma unroll
    for (int kc = 0; kc < NKC; ++kc) {
        int p0 = kc * 4 + koff;
        int o0 = p0 * HSTR + m;
        v2f b;
        b.x = Hb[0 * HROWS * HSTR + o0];
        b.y = Hb[0 * HROWS * HSTR + o0 + HSTR];
        c1 = WMMA_F32(gband[kc], b, c1);
        b.x = Hb[1 * HROWS * HSTR + o0];
        b.y = Hb[1 * HROWS * HSTR + o0 + HSTR];
        c2 = WMMA_F32(gband[kc], b, c2);
        b.x = Hb[2 * HROWS * HSTR + o0];
        b.y = Hb[2 * HROWS * HSTR + o0 + HSTR];
        c11 = WMMA_F32(gband[kc], b, c11);
        b.x = Hb[3 * HROWS * HSTR + o0];
        b.y = Hb[3 * HROWS * HSTR + o0 + HSTR];
        c22 = WMMA_F32(gband[kc], b, c22);
        b.x = Hb[4 * HROWS * HSTR + o0];
        b.y = Hb[4 * HROWS * HSTR + o0 + HSTR];
        c12 = WMMA_F32(gband[kc], b, c12);
    }

    // ---- SSIM map on this lane's 8 output pixels + wave reduction ----
    float lsum = 0.0f;
#pragma unroll
    for (int i = 0; i < 8; ++i) {
        float mu1 = c1[i];
        float mu2 = c2[i];
        float mu1sq = mu1 * mu1;
        float mu2sq = mu2 * mu2;
        float mu12  = mu1 * mu2;
        float s1  = c11[i] - mu1sq;
        float s2  = c22[i] - mu2sq;
        float s12 = c12[i] - mu12;
        float num = (2.0f * mu12 + SSIM_C1) * (2.0f * s12 + SSIM_C2);
        float den = (mu1sq + mu2sq + SSIM_C1) * (s1 + s2 + SSIM_C2);
        lsum += num * __builtin_amdgcn_rcpf(den);   // v_rcp_f32 (~1 ulp): fine for a mean
    }
#pragma unroll
    for (int off = 16; off > 0; off >>= 1)
        lsum += __shfl_xor(lsum, off, 32);

    if (lane == 0 && active)
        atomicAdd(out, lsum * inv_total);
}

extern "C" void kernel_launch(void* const* d_in, const int* in_sizes, int n_in,
                              void* d_out, int out_size, void* d_ws, size_t ws_size,
                              hipStream_t stream) {
    const float* img1 = (const float*)d_in[0];
    const float* img2 = (const float*)d_in[1];
    float* out = (float*)d_out;

    const int N = 32, C = 3;
    const int tiles_per_plane = NTX * NTX;        // 1024
    const int n_tiles = N * C * tiles_per_plane;  // 98304
    const int blocks = (n_tiles + WPB - 1) / WPB;
    const float inv_total = 1.0f / ((float)N * C * IMGDIM * IMGDIM);

    ssim_zero_out<<<1, 32, 0, stream>>>(out);
    ssim_wmma_kernel<<<blocks, WPB * 32, 0, stream>>>(
        img1, img2, out, n_tiles, inv_total);
}